// NCC_67113158968053
// MI455X (gfx1250) — compile-verified
//
#include <hip/hip_runtime.h>

// NCC loss: fused 9x9 box-filter of 5 moment maps + cc + global mean, one pass.
// I = y_true, J = y_pred. B=32, C=4, H=W=512.

#define HH 512
#define WW 512
#define TILE 64
#define HALO 4
#define IN 72              // TILE + 2*HALO
#define SIW 77             // padded LDS stride for input tiles (conflict-free)
#define HSW 73             // padded LDS stride for horizontal-sum buffer
#define NTHREADS 256

typedef __attribute__((ext_vector_type(2))) float v2f;
typedef __attribute__((ext_vector_type(8))) float v8f;

__global__ __launch_bounds__(NTHREADS) void ncc_fused_kernel(
    const float* __restrict__ J,   // y_pred
    const float* __restrict__ I,   // y_true
    float* __restrict__ out,       // single accumulator
    float invN)                    // 1 / (B*C*H*W)
{
    __shared__ float sI[IN * SIW];     // 22176 B
    __shared__ float sJ[IN * SIW];     // 22176 B
    __shared__ float hs[IN * HSW];     // 21024 B (reused per moment)
    __shared__ float red[8];           // 65408 B total

    const int tid  = threadIdx.x;
    const int lane = tid & 31;
    const int wave = tid >> 5;
    const int half = (lane >> 4) & 1;  // 0: lanes 0-15, 1: lanes 16-31
    const int mrow = lane & 15;        // M index in A / N index in B,D

    // ---- tile coordinates ----
    const int tilex = (blockIdx.x & 7) * TILE;
    const int tiley = (blockIdx.x >> 3) * TILE;
    const size_t imgOff = (size_t)blockIdx.y * (size_t)(HH * WW);
    const float* gI = I + imgOff;
    const float* gJ = J + imgOff;

    // ---- Phase 1: stage 72x72 region (zero-padded) into LDS, b128 fast path ----
    // 72 rows x 18 float4-groups; gx = tilex-4+4k is 16B-aligned (tilex % 64 == 0).
    for (int idx = tid; idx < IN * 18; idx += NTHREADS) {
        int r  = idx / 18;
        int c4 = (idx - r * 18) * 4;
        int gy = tiley - HALO + r;
        int gx = tilex - HALO + c4;
        float* dI = &sI[r * SIW + c4];
        float* dJ = &sJ[r * SIW + c4];
        if ((unsigned)gy < (unsigned)HH && gx >= 0 && gx + 3 < WW) {
            const size_t o = (size_t)gy * WW + gx;
            float4 vI = *(const float4*)(gI + o);
            float4 vJ = *(const float4*)(gJ + o);
            dI[0] = vI.x; dI[1] = vI.y; dI[2] = vI.z; dI[3] = vI.w;
            dJ[0] = vJ.x; dJ[1] = vJ.y; dJ[2] = vJ.z; dJ[3] = vJ.w;
        } else {
#pragma unroll
            for (int u = 0; u < 4; ++u) {
                int gxu = gx + u;
                bool in = ((unsigned)gy < (unsigned)HH) && ((unsigned)gxu < (unsigned)WW);
                size_t o = (size_t)gy * WW + gxu;
                dI[u] = in ? gI[o] : 0.0f;
                dJ[u] = in ? gJ[o] : 0.0f;
            }
        }
    }

    // ---- Banded-ones A fragments (16x24 band, K=4 slices) ----
    // A[m][k] = 1 iff m <= k <= m+8. 32-bit A 16x4 layout:
    //   VGPR0 K = 4*s + 2*half, VGPR1 K = 4*s + 2*half + 1, M = lane&15.
    v2f a6[6];
#pragma unroll
    for (int s = 0; s < 6; ++s) {
        int k0 = 4 * s + 2 * half;
        a6[s].x = (k0 >= mrow && k0 <= mrow + 8) ? 1.0f : 0.0f;
        a6[s].y = (k0 + 1 >= mrow && k0 + 1 <= mrow + 8) ? 1.0f : 0.0f;
    }

    v8f acc[5][2];   // 5 moments x 2 subtiles, 16x16 f32 D fragments

    // ---- Phase 2: per moment, running-sum horizontal pass + WMMA vertical ----
#pragma unroll
    for (int m = 0; m < 5; ++m) {
        __syncthreads();   // hs free (prev WMMA reads / load phase done)

        // horizontal 9-tap running sum: 72 rows x 4 quarters of 16 output cols
        for (int idx = tid; idx < IN * 4; idx += NTHREADS) {
            int r = idx >> 2;
            int q = idx & 3;
            const float* rowI = &sI[r * SIW + 16 * q];
            const float* rowJ = &sJ[r * SIW + 16 * q];
            float* hrow = &hs[r * HSW + 16 * q];
            float wbuf[24];
            float s = 0.0f;
#pragma unroll
            for (int t = 0; t < 8; ++t) {
                float iv = rowI[t], jv = rowJ[t];
                float v = (m == 0) ? iv : (m == 1) ? jv : (m == 2) ? iv * iv
                        : (m == 3) ? jv * jv : iv * jv;
                wbuf[t] = v;
                s += v;
            }
#pragma unroll
            for (int x = 0; x < 16; ++x) {
                float iv = rowI[x + 8], jv = rowJ[x + 8];
                float v = (m == 0) ? iv : (m == 1) ? jv : (m == 2) ? iv * iv
                        : (m == 3) ? jv * jv : iv * jv;
                wbuf[x + 8] = v;
                s += v;          // s = sum cols x..x+8
                hrow[x] = s;
                s -= wbuf[x];    // slide window
            }
        }
        __syncthreads();

        // vertical pass: D(16x16) = A(16x24 band) * HS(24x16), 6 chained K=4 WMMAs.
        // 16 subtiles (4x4) over the 64x64 tile; wave w owns subtiles 2w, 2w+1.
#pragma unroll
        for (int st = 0; st < 2; ++st) {
            int sub = 2 * wave + st;
            int ty = sub >> 2;
            int tx = sub & 3;
            int rbase = 16 * ty;
            int cbase = 16 * tx + mrow;   // N = lane&15
            v8f c = {0.f, 0.f, 0.f, 0.f, 0.f, 0.f, 0.f, 0.f};
#pragma unroll
            for (int s = 0; s < 6; ++s) {
                // B 4x16 layout: VGPR0 row = 4*s + 2*half, VGPR1 row = +1
                int kr = rbase + 4 * s + 2 * half;
                v2f b;
                b.x = hs[kr * HSW + cbase];
                b.y = hs[(kr + 1) * HSW + cbase];
                c = __builtin_amdgcn_wmma_f32_16x16x4_f32(
                        false, a6[s], false, b, (short)0, c, false, false);
            }
            acc[m][st] = c;
        }
    }
    __syncthreads();

    // ---- Phase 3: per-pixel cc from the 5 summed moments, then reduce ----
    const float inv_n = 1.0f / 81.0f;
    float local = 0.0f;
#pragma unroll
    for (int st = 0; st < 2; ++st) {
#pragma unroll
        for (int v = 0; v < 8; ++v) {
            float sumI  = acc[0][st][v];
            float sumJ  = acc[1][st][v];
            float sumI2 = acc[2][st][v];
            float sumJ2 = acc[3][st][v];
            float sumIJ = acc[4][st][v];
            float muI = sumI * inv_n;
            float muJ = sumJ * inv_n;
            float varI = sumI2 * inv_n - muI * muI;
            float varJ = sumJ2 * inv_n - muJ * muJ;
            float cov  = sumIJ * inv_n - muI * muJ;
            float den  = fmaxf(varI, 0.0f) * fmaxf(varJ, 0.0f) + 1e-5f;
            local += (cov * cov) / den;
        }
    }

    // wave32 tree reduction
#pragma unroll
    for (int off = 16; off > 0; off >>= 1)
        local += __shfl_down(local, off, 32);
    if (lane == 0) red[wave] = local;
    __syncthreads();
    if (tid == 0) {
        float t = 0.0f;
#pragma unroll
        for (int w = 0; w < 8; ++w) t += red[w];
        atomicAdd(out, -t * invN);
    }
}

extern "C" void kernel_launch(void* const* d_in, const int* in_sizes, int n_in,
                              void* d_out, int out_size, void* d_ws, size_t ws_size,
                              hipStream_t stream) {
    (void)n_in; (void)d_ws; (void)ws_size; (void)out_size;
    const float* y_pred = (const float*)d_in[0];
    const float* y_true = (const float*)d_in[1];
    float* out = (float*)d_out;

    const long long total = (long long)in_sizes[0];          // B*C*H*W
    const int images = (int)(total / (long long)(HH * WW));  // B*C
    const float invN = 1.0f / (float)total;

    hipMemsetAsync(out, 0, sizeof(float), stream);           // capture-safe

    dim3 grid((HH / TILE) * (WW / TILE), images);            // (64, 128)
    dim3 block(NTHREADS);
    ncc_fused_kernel<<<grid, block, 0, stream>>>(y_pred, y_true, out, invN);
}